// ouroboros_69492570849411
// MI455X (gfx1250) — compile-verified
//
#include <hip/hip_runtime.h>
#include <hip/hip_bf16.h>

typedef __attribute__((ext_vector_type(2))) float v2f;
typedef __attribute__((ext_vector_type(8))) float v8f;

// Exact pointee type the async-LDS builtin wants: GCC-style int vector, 16B.
typedef int v4i_vs __attribute__((vector_size(16)));
typedef __attribute__((address_space(1))) v4i_vs as1_v4i;
typedef __attribute__((address_space(3))) v4i_vs as3_v4i;

#define GEMM_KB 32
#define LDS_STRIDE 36   // bank-conflict-free fragment reads AND 16B-aligned rows for async b128

#if defined(__HIP_DEVICE_COMPILE__) && __has_builtin(__builtin_amdgcn_global_load_async_to_lds_b128)
#define HAVE_ASYNC_LDS 1
#else
#define HAVE_ASYNC_LDS 0
#endif

__device__ __forceinline__ float silu_f(float x) {
    return x / (1.f + __expf(-x));
}
__device__ __forceinline__ float softplus_f(float x) {
    return fmaxf(x, 0.f) + log1pf(__expf(-fabsf(x)));
}

__device__ __forceinline__ void wait_async_loads() {
#if HAVE_ASYNC_LDS
#if __has_builtin(__builtin_amdgcn_s_wait_asynccnt)
    __builtin_amdgcn_s_wait_asynccnt(0);
#else
    asm volatile("s_wait_asynccnt 0x0" ::: "memory");
#endif
#endif
}

// ---------------------------------------------------------------------------
// Tile staging: A tile 64 x 32 and W tile 128 x 32 into LDS.
// Fast path: async global->LDS b128 (ASYNCcnt), used when the whole tile is
// in-bounds. Fallback: scalar zero-fill (ragged N / ragged K tiles).
// ---------------------------------------------------------------------------
__device__ __forceinline__ void load_tiles(const float* __restrict__ A, int lda,
                                           const float* __restrict__ W, int ldw,
                                           int m0, int n0, int k0,
                                           int M, int N, int K,
                                           float* bA, float* bW, int tid) {
#if HAVE_ASYNC_LDS
    bool full = (m0 + 64 <= M) && (n0 + 128 <= N) && (k0 + GEMM_KB <= K);
    if (full) {
#pragma unroll
        for (int it = 0; it < 2; ++it) {           // A: 512 float4 / 256 thr
            int i = tid + it * 256;
            int r = i >> 3, c4 = i & 7;
            const float* gp = A + (size_t)(m0 + r) * lda + (k0 + c4 * 4);
            float* lp = bA + r * LDS_STRIDE + c4 * 4;
            __builtin_amdgcn_global_load_async_to_lds_b128(
                (as1_v4i*)gp, (as3_v4i*)lp, 0, 0);
        }
#pragma unroll
        for (int it = 0; it < 4; ++it) {           // W: 1024 float4 / 256 thr
            int i = tid + it * 256;
            int r = i >> 3, c4 = i & 7;
            const float* gp = W + (size_t)(n0 + r) * ldw + (k0 + c4 * 4);
            float* lp = bW + r * LDS_STRIDE + c4 * 4;
            __builtin_amdgcn_global_load_async_to_lds_b128(
                (as1_v4i*)gp, (as3_v4i*)lp, 0, 0);
        }
        return;
    }
#endif
    // zero-fill scalar fallback (edge tiles)
    for (int i = tid; i < 64 * GEMM_KB; i += 256) {
        int r = i >> 5, c = i & 31;
        int gr = m0 + r, gk = k0 + c;
        bA[r * LDS_STRIDE + c] = (gr < M && gk < K) ? A[(size_t)gr * lda + gk] : 0.f;
    }
    for (int i = tid; i < 128 * GEMM_KB; i += 256) {
        int r = i >> 5, c = i & 31;
        int gn = n0 + r, gk = k0 + c;
        bW[r * LDS_STRIDE + c] = (gn < N && gk < K) ? W[(size_t)gn * ldw + gk] : 0.f;
    }
}

// ---------------------------------------------------------------------------
// Generic fp32 WMMA GEMM:  C[M,N] = A[M,K] * W[N,K]^T  (+bias[N]) (softplus) (+add[M,N])
// block = 256 threads (8 waves). Block tile 64(M) x 128(N). Wave tile 16 x 64.
// Double-buffered LDS, async loads pipelined against WMMA.
// ---------------------------------------------------------------------------
__device__ __forceinline__ void gemm_epilogue(v8f acc, int gcol, int rbase,
                                              int M, int N,
                                              const float* __restrict__ bias,
                                              const float* __restrict__ addsrc, int ldadd,
                                              float* __restrict__ C, int ldc, int act) {
    if (gcol >= N) return;
    float bv = bias ? bias[gcol] : 0.f;
#pragma unroll
    for (int i = 0; i < 8; ++i) {
        int grow = rbase + i;
        if (grow < M) {
            float v = acc[i] + bv;
            if (act == 1) v = softplus_f(v);
            if (addsrc) v += addsrc[(size_t)grow * ldadd + gcol];
            C[(size_t)grow * ldc + gcol] = v;
        }
    }
}

__global__ __launch_bounds__(256) void gemm_wmma_kernel(
    const float* __restrict__ A, int lda,
    const float* __restrict__ W, int ldw,
    float* __restrict__ C, int ldc,
    const float* __restrict__ bias,
    const float* __restrict__ addsrc, int ldadd,
    int M, int N, int K, int act) {

    __shared__ __align__(16) float sA[2][64 * LDS_STRIDE];
    __shared__ __align__(16) float sW[2][128 * LDS_STRIDE];

    const int tid   = threadIdx.x;
    const int lane  = tid & 31;
    const int wid   = tid >> 5;     // 0..7
    const int wm    = wid >> 1;     // 0..3 : M sub-tile
    const int wn    = wid & 1;      // 0..1 : N sub-tile (64 wide)
    const int m0    = blockIdx.x * 64;
    const int n0    = blockIdx.y * 128;
    const int row16 = lane & 15;
    const int khalf = (lane < 16) ? 0 : 2;

    v8f acc0 = {}, acc1 = {}, acc2 = {}, acc3 = {};

    const int nchunks = (K + GEMM_KB - 1) / GEMM_KB;

    // prologue: stage chunk 0 into buffer 0
    load_tiles(A, lda, W, ldw, m0, n0, 0, M, N, K, sA[0], sW[0], tid);
    wait_async_loads();
    __syncthreads();

    int p = 0;
    for (int ch = 0; ch < nchunks; ++ch) {
        // issue next chunk's loads into the other buffer (overlaps with WMMA below)
        if (ch + 1 < nchunks)
            load_tiles(A, lda, W, ldw, m0, n0, (ch + 1) * GEMM_KB, M, N, K,
                       sA[p ^ 1], sW[p ^ 1], tid);

        const float* bA = sA[p];
        const float* bW = sW[p];
#pragma unroll
        for (int kk = 0; kk < GEMM_KB; kk += 4) {
            v2f a  = *(const v2f*)&bA[(wm * 16 + row16) * LDS_STRIDE + kk + khalf];
            v2f b0 = *(const v2f*)&bW[(wn * 64 +  0 + row16) * LDS_STRIDE + kk + khalf];
            v2f b1 = *(const v2f*)&bW[(wn * 64 + 16 + row16) * LDS_STRIDE + kk + khalf];
            v2f b2 = *(const v2f*)&bW[(wn * 64 + 32 + row16) * LDS_STRIDE + kk + khalf];
            v2f b3 = *(const v2f*)&bW[(wn * 64 + 48 + row16) * LDS_STRIDE + kk + khalf];
            acc0 = __builtin_amdgcn_wmma_f32_16x16x4_f32(false, a, false, b0, (short)0, acc0, false, false);
            acc1 = __builtin_amdgcn_wmma_f32_16x16x4_f32(false, a, false, b1, (short)0, acc1, false, false);
            acc2 = __builtin_amdgcn_wmma_f32_16x16x4_f32(false, a, false, b2, (short)0, acc2, false, false);
            acc3 = __builtin_amdgcn_wmma_f32_16x16x4_f32(false, a, false, b3, (short)0, acc3, false, false);
        }
        wait_async_loads();
        __syncthreads();
        p ^= 1;
    }

    const int rbase = m0 + wm * 16 + ((lane < 16) ? 0 : 8);
    const int cb    = n0 + wn * 64 + row16;
    gemm_epilogue(acc0, cb +  0, rbase, M, N, bias, addsrc, ldadd, C, ldc, act);
    gemm_epilogue(acc1, cb + 16, rbase, M, N, bias, addsrc, ldadd, C, ldc, act);
    gemm_epilogue(acc2, cb + 32, rbase, M, N, bias, addsrc, ldadd, C, ldc, act);
    gemm_epilogue(acc3, cb + 48, rbase, M, N, bias, addsrc, ldadd, C, ldc, act);
}

// ---------------------------------------------------------------------------
// x_in = concat([y-x, y], -1) then concat([x_in, flip_L(x_in)], -1): (B,L,256)
// ---------------------------------------------------------------------------
__global__ void build_xin_kernel(const float* __restrict__ x, const float* __restrict__ y,
                                 float* __restrict__ xin, int total) {
    int t = blockIdx.x * blockDim.x + threadIdx.x;
    if (t >= total) return;
    int c   = t & 255;
    int row = t >> 8;              // b*1024 + l
    int l   = row & 1023;
    int b   = row >> 10;
    int src = (c < 128) ? row : (b * 1024 + (1023 - l));
    int cc  = (c < 128) ? c : (c - 128);
    float v;
    if (cc < 64) v = y[(size_t)src * 64 + cc] - x[(size_t)src * 64 + cc];
    else         v = y[(size_t)src * 64 + (cc - 64)];
    xin[t] = v;
}

// ---------------------------------------------------------------------------
// RMSNorm, one block per (b,l) row
// ---------------------------------------------------------------------------
__global__ __launch_bounds__(128) void rmsnorm_kernel(const float* __restrict__ x,
                                                      const float* __restrict__ w,
                                                      float* __restrict__ out, int D) {
    __shared__ float red[128];
    int row = blockIdx.x;
    int tid = threadIdx.x;
    float s = 0.f;
    for (int i = tid; i < D; i += 128) {
        float v = x[(size_t)row * D + i];
        s += v * v;
    }
    red[tid] = s;
    __syncthreads();
    for (int off = 64; off > 0; off >>= 1) {
        if (tid < off) red[tid] += red[tid + off];
        __syncthreads();
    }
    float scale = rsqrtf(red[0] / (float)D + 1e-5f);
    for (int i = tid; i < D; i += 128)
        out[(size_t)row * D + i] = x[(size_t)row * D + i] * scale * w[i];
}

// ---------------------------------------------------------------------------
// causal depthwise conv (width 4) + bias + SiLU. input = first half of xz.
// ---------------------------------------------------------------------------
__global__ void conv_silu_kernel(const float* __restrict__ xz, int ldxz,
                                 const float* __restrict__ cw, const float* __restrict__ cb,
                                 float* __restrict__ xh, int total, int ED, int L) {
    int t = blockIdx.x * blockDim.x + threadIdx.x;
    if (t >= total) return;
    int e   = t % ED;
    int row = t / ED;
    int l   = row % L;
    int b   = row / L;
    float acc = cb[e];
#pragma unroll
    for (int j = 0; j < 4; ++j) {
        int ll = l - 3 + j;
        if (ll >= 0)
            acc += cw[e * 4 + j] * xz[(size_t)(b * L + ll) * ldxz + e];
    }
    xh[t] = silu_f(acc);
}

// ---------------------------------------------------------------------------
// selective scan: one thread per (b, channel), N=16 state in registers
// ---------------------------------------------------------------------------
__global__ void scan_kernel(const float* __restrict__ xh, const float* __restrict__ delta,
                            const float* __restrict__ dbc, int dbc_ld, int dt_rank,
                            const float* __restrict__ A_log, const float* __restrict__ Dp,
                            float* __restrict__ yout, int total, int ED) {
    int t = blockIdx.x * blockDim.x + threadIdx.x;
    if (t >= total) return;
    int e = t % ED;
    int b = t / ED;
    float Ac[16];
#pragma unroll
    for (int n = 0; n < 16; ++n) Ac[n] = -__expf(A_log[e * 16 + n]);
    float Dv = Dp[e];
    float h[16];
#pragma unroll
    for (int n = 0; n < 16; ++n) h[n] = 0.f;
    const int L = 1024;
    for (int l = 0; l < L; ++l) {
        size_t row = (size_t)b * L + l;
        float d  = delta[row * ED + e];
        float xv = xh[row * ED + e];
        float dx = d * xv;
        const float* bc = dbc + row * dbc_ld + dt_rank;
        float y = 0.f;
#pragma unroll
        for (int n = 0; n < 16; ++n) {
            float dA = __expf(d * Ac[n]);
            h[n] = dA * h[n] + dx * bc[n];
            y += h[n] * bc[16 + n];
        }
        yout[row * ED + e] = y + Dv * xv;
    }
}

// ---------------------------------------------------------------------------
// y *= silu(z), z = second half of xz
// ---------------------------------------------------------------------------
__global__ void gate_kernel(float* __restrict__ y, const float* __restrict__ xz,
                            int total, int ED) {
    int t = blockIdx.x * blockDim.x + threadIdx.x;
    if (t >= total) return;
    int e   = t % ED;
    int row = t / ED;
    float zv = xz[(size_t)row * (2 * ED) + ED + e];
    y[t] *= silu_f(zv);
}

// ---------------------------------------------------------------------------
// state_pred = flip_L(silu(sp))  (D = 512)
// ---------------------------------------------------------------------------
__global__ void silu_flip_kernel(const float* __restrict__ sp, float* __restrict__ out,
                                 int total) {
    int t = blockIdx.x * blockDim.x + threadIdx.x;
    if (t >= total) return;
    int c   = t & 511;
    int row = t >> 9;
    int l   = row & 1023;
    int b   = row >> 10;
    float v = sp[((size_t)(b * 1024 + (1023 - l))) * 512 + c];
    out[t] = silu_f(v);
}

// ===========================================================================
// host side
// ===========================================================================
struct LayerP {
    const float *norm_w, *in_proj, *conv_w, *conv_b, *x_proj, *dt_w, *dt_b, *A_log, *D, *out_proj;
};
static LayerP get_layer(void* const* d_in, int base) {
    LayerP p;
    p.norm_w   = (const float*)d_in[base + 0];
    p.in_proj  = (const float*)d_in[base + 1];
    p.conv_w   = (const float*)d_in[base + 2];
    p.conv_b   = (const float*)d_in[base + 3];
    p.x_proj   = (const float*)d_in[base + 4];
    p.dt_w     = (const float*)d_in[base + 5];
    p.dt_b     = (const float*)d_in[base + 6];
    p.A_log    = (const float*)d_in[base + 7];
    p.D        = (const float*)d_in[base + 8];
    p.out_proj = (const float*)d_in[base + 9];
    return p;
}

struct Scratch {
    float *xin, *cx0, *cx1, *xn, *xz, *xh, *dbc, *delta, *y;
};

static void launch_gemm(const float* A, int lda, const float* W, int ldw,
                        float* C, int ldc, const float* bias,
                        const float* addsrc, int ldadd,
                        int M, int N, int K, int act, hipStream_t s) {
    dim3 grid((M + 63) / 64, (N + 127) / 128);
    gemm_wmma_kernel<<<grid, 256, 0, s>>>(A, lda, W, ldw, C, ldc, bias, addsrc, ldadd, M, N, K, act);
}

static void run_mamba_block(const LayerP& p, int Dm, int dt_rank,
                            const float* xin, float* xout, const Scratch& w, hipStream_t s) {
    const int M = 4096;            // B*L
    const int ED = Dm;             // expand = 1
    const int dbc_ld = dt_rank + 32;
    const int tot = M * ED;

    rmsnorm_kernel<<<M, 128, 0, s>>>(xin, p.norm_w, w.xn, Dm);
    launch_gemm(w.xn, Dm, p.in_proj, Dm, w.xz, 2 * ED, nullptr, nullptr, 0, M, 2 * ED, Dm, 0, s);
    conv_silu_kernel<<<(tot + 255) / 256, 256, 0, s>>>(w.xz, 2 * ED, p.conv_w, p.conv_b, w.xh, tot, ED, 1024);
    launch_gemm(w.xh, ED, p.x_proj, ED, w.dbc, dbc_ld, nullptr, nullptr, 0, M, dbc_ld, ED, 0, s);
    // delta = softplus(dbc[:, :dt_rank] @ dt_w^T + dt_b)
    launch_gemm(w.dbc, dbc_ld, p.dt_w, dt_rank, w.delta, ED, p.dt_b, nullptr, 0, M, ED, dt_rank, 1, s);
    int ns = 4 * ED;               // B * ED
    scan_kernel<<<(ns + 255) / 256, 256, 0, s>>>(w.xh, w.delta, w.dbc, dbc_ld, dt_rank,
                                                 p.A_log, p.D, w.y, ns, ED);
    gate_kernel<<<(tot + 255) / 256, 256, 0, s>>>(w.y, w.xz, tot, ED);
    launch_gemm(w.y, ED, p.out_proj, ED, xout, Dm, nullptr, xin, Dm, M, Dm, ED, 0, s);
}

extern "C" void kernel_launch(void* const* d_in, const int* in_sizes, int n_in,
                              void* d_out, int out_size, void* d_ws, size_t ws_size,
                              hipStream_t stream) {
    const float* x = (const float*)d_in[0];
    const float* y = (const float*)d_in[1];
    LayerP c0 = get_layer(d_in, 2);
    LayerP c1 = get_layer(d_in, 12);
    LayerP g0 = get_layer(d_in, 22);
    LayerP g1 = get_layer(d_in, 32);
    const float* cp_w = (const float*)d_in[42];
    const float* cp_b = (const float*)d_in[43];
    const float* op_w = (const float*)d_in[44];
    const float* op_b = (const float*)d_in[45];

    float* ws = (float*)d_ws;
    size_t o = 0;
    Scratch sc;
    sc.xin   = ws + o; o += (size_t)4096 * 256;   // also reused for data-stream ping/pong
    sc.cx0   = ws + o; o += (size_t)4096 * 512;
    sc.cx1   = ws + o; o += (size_t)4096 * 512;
    sc.xn    = ws + o; o += (size_t)4096 * 512;
    sc.xz    = ws + o; o += (size_t)4096 * 1024;
    sc.xh    = ws + o; o += (size_t)4096 * 512;
    sc.dbc   = ws + o; o += (size_t)4096 * 96;
    sc.delta = ws + o; o += (size_t)4096 * 512;
    sc.y     = ws + o; o += (size_t)4096 * 512;

    float* yhat_out  = (float*)d_out;               // (B,L,32)   = 131072
    float* state_out = (float*)d_out + 131072;      // (B,L,512)  = 2097152
    float* dx0 = sc.xin;                            // data stream ping/pong (each 131072)
    float* dx1 = sc.xin + 131072;

    // 1) x_in
    {
        int total = 4096 * 256;
        build_xin_kernel<<<(total + 255) / 256, 256, 0, stream>>>(x, y, sc.xin, total);
    }
    // 2) u = x_in @ cp_w^T + cp_b   (xin consumed here before dx0/dx1 aliasing matters)
    launch_gemm(sc.xin, 256, cp_w, 256, sc.cx0, 512, cp_b, nullptr, 0, 4096, 512, 256, 0, stream);
    // 3) control mamba x2 (residual handled in out_proj epilogue)
    run_mamba_block(c0, 512, 32, sc.cx0, sc.cx1, sc, stream);
    run_mamba_block(c1, 512, 32, sc.cx1, sc.cx0, sc, stream);
    // 4) state_pred = flip(silu(.))  -> second output region
    {
        int total = 4096 * 512;
        silu_flip_kernel<<<(total + 255) / 256, 256, 0, stream>>>(sc.cx0, state_out, total);
    }
    // 5) yhat0 = state_pred @ op_w^T + op_b
    launch_gemm(state_out, 512, op_w, 512, dx0, 32, op_b, nullptr, 0, 4096, 32, 512, 0, stream);
    // 6) data mamba x2, final block writes directly into d_out yhat region
    run_mamba_block(g0, 32, 2, dx0, dx1, sc, stream);
    run_mamba_block(g1, 32, 2, dx1, yhat_out, sc, stream);
}